// VectorToOffsetMatrix_81509889343749
// MI455X (gfx1250) — compile-verified
//
#include <hip/hip_runtime.h>

#define DIM   1024
#define BATCH 128
#define ROWS_PER_BLOCK 64
#define THREADS 256

typedef float v4f  __attribute__((ext_vector_type(4)));
typedef int   i32x4 __attribute__((ext_vector_type(4)));

// gfx1250 async global->LDS copy path (ASYNCcnt-tracked), with safe fallback.
#if defined(__HIP_DEVICE_COMPILE__) && defined(__gfx1250__) && \
    __has_builtin(__builtin_amdgcn_global_load_async_to_lds_b128) && \
    __has_builtin(__builtin_amdgcn_s_wait_asynccnt)
#define USE_ASYNC_LDS 1
#define GLOBAL_AS __attribute__((address_space(1)))
#define LDS_AS    __attribute__((address_space(3)))
#endif

// out[b, i, j] = x[b, (j - i) & 1023]
// One workgroup: one batch b, ROWS_PER_BLOCK consecutive rows i.
// x[b,:] staged in LDS (duplicated 2x so a 4-wide read never wraps).
// Thread t owns destination columns 4t..4t+3 -> aligned, coalesced b128 NT stores.
__global__ __launch_bounds__(THREADS) void circulant_rows_kernel(
    const float* __restrict__ x, float* __restrict__ out)
{
    __shared__ float srow[2 * DIM];  // 8 KB

    const int b    = blockIdx.x;
    const int row0 = blockIdx.y * ROWS_PER_BLOCK;
    const int tid  = threadIdx.x;

    const float* __restrict__ xrow = x + (size_t)b * DIM;

#if defined(USE_ASYNC_LDS)
    {
        const int e0 = tid * 4;  // 256 threads x 4 floats = full 1024-row
        GLOBAL_AS i32x4* g  = (GLOBAL_AS i32x4*)(xrow + e0);
        LDS_AS    i32x4* l0 = (LDS_AS    i32x4*)&srow[e0];
        LDS_AS    i32x4* l1 = (LDS_AS    i32x4*)&srow[DIM + e0];
        __builtin_amdgcn_global_load_async_to_lds_b128(g, l0, 0, 0);
        __builtin_amdgcn_global_load_async_to_lds_b128(g, l1, 0, 0);
        __builtin_amdgcn_s_wait_asynccnt(0);
    }
#else
    for (int t = tid; t < DIM; t += THREADS) {
        float v = xrow[t];
        srow[t]       = v;
        srow[t + DIM] = v;
    }
#endif
    __syncthreads();

    const int j0 = tid * 4;
    float* __restrict__ obase =
        out + (size_t)b * DIM * DIM + (size_t)row0 * DIM + j0;

#pragma unroll 8
    for (int r = 0; r < ROWS_PER_BLOCK; ++r) {
        const int i = row0 + r;
        const int s = (j0 - i) & (DIM - 1);  // 0..1023; s+3 < 2048 (no wrap)
        v4f v;
        v.x = srow[s + 0];
        v.y = srow[s + 1];
        v.z = srow[s + 2];
        v.w = srow[s + 3];
        __builtin_nontemporal_store(v, (v4f*)(obase + (size_t)r * DIM));
    }
}

extern "C" void kernel_launch(void* const* d_in, const int* in_sizes, int n_in,
                              void* d_out, int out_size, void* d_ws, size_t ws_size,
                              hipStream_t stream)
{
    const float* x = (const float*)d_in[0];
    float* out     = (float*)d_out;
    (void)in_sizes; (void)n_in; (void)out_size; (void)d_ws; (void)ws_size;

    dim3 grid(BATCH, DIM / ROWS_PER_BLOCK);  // (128, 16)
    circulant_rows_kernel<<<grid, THREADS, 0, stream>>>(x, out);
}